// TopologicalLoss_6777458393524
// MI455X (gfx1250) — compile-verified
//
#include <hip/hip_runtime.h>

// ---------------------------------------------------------------------------
// TopologicalLoss on MI455X (gfx1250, wave32)
//   out = (1/8) * sum_{i,j} m[i,j] * sum_{8 neighbors n} sigmoid(10*(1-S8+n))
// with S8 = sum of 8-neighborhood, out-of-image value = 1.0.
// Transcendental-minimized: F = exp(-10 m) precomputed per element,
// r = exp(10(S8-1)) per pixel, pairwise rational combine -> 1 exp2 + 4 rcp /px.
// Wave reduction via V_WMMA_F32_16X16X4_F32 (f32 -> exact).
// ---------------------------------------------------------------------------

typedef __attribute__((ext_vector_type(2))) float v2f;
typedef __attribute__((ext_vector_type(8))) float v8f;

#define IMG 4096
#define TILE 64
#define LDS_STRIDE 68   // 66 used cols, padded stride
#define L10 14.426950408889634f   // 10 * log2(e)

__device__ __forceinline__ float sigpair(float a, float b) {
    // 1/(1+a) + 1/(1+b) = (2+a+b) / (1+a+b+ab); ab overflow -> rcp(inf)=0 (ok)
    float s   = a + b;
    float num = s + 2.0f;
    float den = __builtin_fmaf(a, b, 1.0f) + s;
    return num * __builtin_amdgcn_rcpf(den);
}

__global__ __launch_bounds__(256)
void topo_tile_kernel(const float* __restrict__ mask, float* __restrict__ partials) {
    __shared__ float sM[66 * LDS_STRIDE];
    __shared__ float sF[66 * LDS_STRIDE];
    __shared__ float sWave[8];

    const int tid = threadIdx.x;
    const int bx = blockIdx.x, by = blockIdx.y;

    // ---- stage tile + halo into LDS; precompute F = exp(-10*m) = exp2(-L10*m)
    for (int idx = tid; idx < 66 * 66; idx += 256) {
        int lr = idx / 66;
        int lc = idx - lr * 66;
        int gy = by * TILE + lr - 1;
        int gx = bx * TILE + lc - 1;
        float v = 1.0f;  // pad value
        if ((unsigned)gy < (unsigned)IMG && (unsigned)gx < (unsigned)IMG)
            v = mask[gy * IMG + gx];
        sM[lr * LDS_STRIDE + lc] = v;
        sF[lr * LDS_STRIDE + lc] = __builtin_amdgcn_exp2f(-L10 * v);
    }
    __syncthreads();

    // ---- each thread computes a 4x4 micro-tile; window is 6x6 in LDS
    const int tx = tid & 15;
    const int ty = tid >> 4;
    const int px = tx * 4;   // multiple of 4 -> 16B-aligned LDS vector loads
    const int py = ty * 4;

    float m[6][6], f[6][6];
#pragma unroll
    for (int r = 0; r < 6; ++r) {
        const float* pm = &sM[(py + r) * LDS_STRIDE + px];
        const float* pf = &sF[(py + r) * LDS_STRIDE + px];
        float4 a4 = *(const float4*)pm;
        float2 a2 = *(const float2*)(pm + 4);
        float4 b4 = *(const float4*)pf;
        float2 b2 = *(const float2*)(pf + 4);
        m[r][0] = a4.x; m[r][1] = a4.y; m[r][2] = a4.z; m[r][3] = a4.w;
        m[r][4] = a2.x; m[r][5] = a2.y;
        f[r][0] = b4.x; f[r][1] = b4.y; f[r][2] = b4.z; f[r][3] = b4.w;
        f[r][4] = b2.x; f[r][5] = b2.y;
    }

    float acc = 0.0f;
#pragma unroll
    for (int r0 = 0; r0 < 4; ++r0) {
        float cs[6];
#pragma unroll
        for (int j = 0; j < 6; ++j)
            cs[j] = m[r0][j] + m[r0 + 1][j] + m[r0 + 2][j];
#pragma unroll
        for (int c0 = 0; c0 < 4; ++c0) {
            float mc  = m[r0 + 1][c0 + 1];
            float box = cs[c0] + cs[c0 + 1] + cs[c0 + 2];
            float S8  = box - mc;
            // r = exp(10*(S8-1)) = exp2(S8*L10 - L10)
            float rt = __builtin_amdgcn_exp2f(__builtin_fmaf(S8, L10, -L10));
            float a0 = rt * f[r0    ][c0    ];
            float a1 = rt * f[r0    ][c0 + 1];
            float a2 = rt * f[r0    ][c0 + 2];
            float a3 = rt * f[r0 + 1][c0    ];
            float a4 = rt * f[r0 + 1][c0 + 2];
            float a5 = rt * f[r0 + 2][c0    ];
            float a6 = rt * f[r0 + 2][c0 + 1];
            float a7 = rt * f[r0 + 2][c0 + 2];
            float sig = sigpair(a0, a1) + sigpair(a2, a3)
                      + sigpair(a4, a5) + sigpair(a6, a7);
            acc = __builtin_fmaf(mc, sig, acc);
        }
    }

    // ---- wave32 reduction via V_WMMA_F32_16X16X4_F32 (exact f32 math)
    // A[m,k]: lane m (<16) VGPR0->K0, VGPR1->K1; lane m+16 VGPR0->K2, VGPR1->K3
    // A = {acc, 0}, B = ones(4x16)  =>  D[m,n] = acc_m + acc_{m+16}
    v2f A; A.x = acc;  A.y = 0.0f;
    v2f B; B.x = 1.0f; B.y = 1.0f;
    v8f C = {};
    v8f D = __builtin_amdgcn_wmma_f32_16x16x4_f32(
        /*neg_a=*/false, A, /*neg_b=*/false, B,
        /*c_mod=*/(short)0, C, /*reuse_a=*/false, /*reuse_b=*/false);
    // lanes 0-15 hold rowsums 0..7 in D[0..7]; lanes 16-31 hold rowsums 8..15
    float t = D[0] + D[1] + D[2] + D[3] + D[4] + D[5] + D[6] + D[7];
    t += __shfl_xor(t, 16, 32);          // full 32-lane wave sum, all lanes

    if ((tid & 31) == 0) sWave[tid >> 5] = t;
    __syncthreads();
    if (tid == 0) {
        float bs = 0.0f;
#pragma unroll
        for (int w = 0; w < 8; ++w) bs += sWave[w];
        partials[by * (IMG / TILE) + bx] = bs;
    }
}

__global__ __launch_bounds__(256)
void topo_reduce_kernel(const float* __restrict__ partials, float* __restrict__ out) {
    __shared__ float s[256];
    const int tid = threadIdx.x;
    float v = 0.0f;
    for (int i = tid; i < (IMG / TILE) * (IMG / TILE); i += 256)
        v += partials[i];           // fixed order -> deterministic
    s[tid] = v;
    __syncthreads();
    for (int off = 128; off > 0; off >>= 1) {
        if (tid < off) s[tid] += s[tid + off];
        __syncthreads();
    }
    if (tid == 0) out[0] = s[0] * 0.125f;   // divide by 8 kernels
}

extern "C" void kernel_launch(void* const* d_in, const int* in_sizes, int n_in,
                              void* d_out, int out_size, void* d_ws, size_t ws_size,
                              hipStream_t stream) {
    const float* mask = (const float*)d_in[0];
    float* out = (float*)d_out;
    float* ws  = (float*)d_ws;   // 64*64 = 4096 block partials (16 KB)

    dim3 grid(IMG / TILE, IMG / TILE);
    topo_tile_kernel<<<grid, 256, 0, stream>>>(mask, ws);
    topo_reduce_kernel<<<1, 256, 0, stream>>>(ws, out);
}